// GraphCorrectionModel_19456201850959
// MI455X (gfx1250) — compile-verified
//
#include <hip/hip_runtime.h>

typedef float v2f __attribute__((ext_vector_type(2)));
typedef float v8f __attribute__((ext_vector_type(8)));

#define HID 64

// ---------------------------------------------------------------------------
// Degree / normalization
// ---------------------------------------------------------------------------
__global__ void k_init_deg(float* __restrict__ deg, int N) {
    int i = blockIdx.x * blockDim.x + threadIdx.x;
    if (i < N) deg[i] = 1.0f;  // self-loop contribution
}

__global__ void k_deg_edges(float* __restrict__ deg, const int* __restrict__ dst, int E) {
    int e = blockIdx.x * blockDim.x + threadIdx.x;
    if (e < E) atomicAdd(&deg[dst[e]], 1.0f);
}

__global__ void k_deg_to_dinv(float* __restrict__ dinv, int N) {
    int i = blockIdx.x * blockDim.x + threadIdx.x;
    if (i < N) {
        float d = dinv[i];
        dinv[i] = (d > 0.0f) ? rsqrtf(d) : 0.0f;
    }
}

// ---------------------------------------------------------------------------
// Layer 1: h = [x, delta] @ W1, pre-scaled by dinv (delta cols are constant)
// ---------------------------------------------------------------------------
__global__ void k_l1_hws(const float* __restrict__ x, const float* __restrict__ delta,
                         const float* __restrict__ W1, const float* __restrict__ dinv,
                         float* __restrict__ hws, int N) {
    int idx = blockIdx.x * blockDim.x + threadIdx.x;
    if (idx >= N * HID) return;
    int n = idx >> 6;
    int c = idx & 63;
    float v = x[n] * W1[c]
            + delta[0] * W1[1 * HID + c]
            + delta[1] * W1[2 * HID + c]
            + delta[2] * W1[3 * HID + c]
            + delta[3] * W1[4 * HID + c];
    hws[idx] = v * dinv[n];
}

// ---------------------------------------------------------------------------
// Edge scatter: acc[dst] += hws[src]   (64-wide rows, float4 granularity)
// ---------------------------------------------------------------------------
__global__ void k_scatter64(const float* __restrict__ hws, const int* __restrict__ src,
                            const int* __restrict__ dst, float* __restrict__ acc, int E) {
    long long idx = (long long)blockIdx.x * blockDim.x + threadIdx.x;
    if (idx >= (long long)E * 16) return;
    int e  = (int)(idx >> 4);
    int c4 = ((int)idx & 15) << 2;
    int s = src[e];
    int d = dst[e];
    const float4 v = *(const float4*)(hws + (size_t)s * HID + c4);
    float* a = acc + (size_t)d * HID + c4;
    atomicAdd(a + 0, v.x);
    atomicAdd(a + 1, v.y);
    atomicAdd(a + 2, v.z);
    atomicAdd(a + 3, v.w);
}

// ---------------------------------------------------------------------------
// Finalize layer: h = relu(dinv * (acc + hws) + b)   (self-loop folded in)
// ---------------------------------------------------------------------------
__global__ void k_finalize64(const float* __restrict__ acc, const float* __restrict__ hws,
                             const float* __restrict__ dinv, const float* __restrict__ b,
                             float* __restrict__ h, int N, int do_relu) {
    int idx = blockIdx.x * blockDim.x + threadIdx.x;
    if (idx >= N * HID) return;
    int n = idx >> 6;
    int c = idx & 63;
    float v = dinv[n] * (acc[idx] + hws[idx]) + b[c];
    if (do_relu) v = fmaxf(v, 0.0f);
    h[idx] = v;
}

// ---------------------------------------------------------------------------
// Layer 2 GEMM via f32 WMMA: hws = (H[N,64] @ W[64,64]) * dinv[row]
// One wave per 16-row tile; 4 column tiles x 16 K-steps of 16x16x4 f32 WMMA.
// A-frag layout (ISA 7.12.2): lane<16 -> K=4kk+{0,1}, lane>=16 -> K=4kk+{2,3}.
// B-frag: VGPR r holds rows K=r (lanes 0-15) and K=r+2 (lanes 16-31).
// C/D:    VGPR r holds row M=r (lanes 0-15) / M=r+8 (lanes 16-31).
// ---------------------------------------------------------------------------
__global__ void k_gemm64_wmma(const float* __restrict__ H, const float* __restrict__ W,
                              const float* __restrict__ dinv, float* __restrict__ out,
                              int N) {
    const int lane = threadIdx.x & 31;
    const int wave = threadIdx.x >> 5;
    const int tile = blockIdx.x * 8 + wave;   // 16-row tile index (wave-uniform)
    const int row0 = tile * 16;
    if (row0 >= N) return;                    // wave-uniform exit; EXEC stays all-1s

    const int half = lane >> 4;               // 0: lanes 0-15, 1: lanes 16-31
    const int l16  = lane & 15;
    const bool full_tile = (row0 + 16 <= N);  // wave-uniform

    // Clamp row for safe A loads on a (theoretical) partial tail tile.
    const int rowA = min(row0 + l16, N - 1);
    const float* arow = H + (size_t)rowA * HID + half * 2;

    // Preload all 16 A fragments for this row tile (reused across 4 col tiles).
    v2f afrag[16];
#pragma unroll
    for (int kk = 0; kk < 16; ++kk) {
        afrag[kk] = *(const v2f*)(arow + kk * 4);   // contiguous 8B load
    }

    // Preload the 8 dinv scale factors this lane's output rows need
    // (rows row0 + r + half*8, r = 0..7; independent of the column tile).
    float dv[8];
#pragma unroll
    for (int r = 0; r < 8; ++r) {
        dv[r] = dinv[min(row0 + r + half * 8, N - 1)];
    }

#pragma unroll
    for (int nt = 0; nt < 4; ++nt) {
        const int n0 = nt * 16;
        v8f c = {};
#pragma unroll
        for (int kk = 0; kk < 16; ++kk) {
            const float* bp = W + (size_t)(kk * 4 + half * 2) * HID + n0 + l16;
            v2f b;
            b.x = bp[0];
            b.y = bp[HID];
            c = __builtin_amdgcn_wmma_f32_16x16x4_f32(
                    /*neg_a=*/false, afrag[kk],
                    /*neg_b=*/false, b,
                    /*c_mod=*/(short)0, c,
                    /*reuse_a=*/false, /*reuse_b=*/false);
        }
        // Store with dinv pre-scaling (producing hws for the edge scatter).
        if (full_tile) {
            // Wave-uniform fast path: straight-line, unconditional stores.
#pragma unroll
            for (int r = 0; r < 8; ++r) {
                const int row = row0 + r + half * 8;
                out[(size_t)row * HID + n0 + l16] = c[r] * dv[r];
            }
        } else {
            // Generic tail tile (not taken when N % 16 == 0).
#pragma unroll
            for (int r = 0; r < 8; ++r) {
                const int row = row0 + r + half * 8;
                if (row < N) {
                    out[(size_t)row * HID + n0 + l16] = c[r] * dv[r];
                }
            }
        }
    }
}

// ---------------------------------------------------------------------------
// Layer 3: 64 -> 3 dot products, pre-scaled by dinv
// ---------------------------------------------------------------------------
__global__ void k_l3_hws(const float* __restrict__ h, const float* __restrict__ W3,
                         const float* __restrict__ dinv, float* __restrict__ hws3, int N) {
    int idx = blockIdx.x * blockDim.x + threadIdx.x;
    if (idx >= N * 3) return;
    int n = idx / 3;
    int c = idx - n * 3;
    const float* hr = h + (size_t)n * HID;
    float s = 0.0f;
#pragma unroll
    for (int k = 0; k < HID; ++k) s += hr[k] * W3[k * 3 + c];
    hws3[idx] = s * dinv[n];
}

__global__ void k_scatter3(const float* __restrict__ hws3, const int* __restrict__ src,
                           const int* __restrict__ dst, float* __restrict__ acc3, int E) {
    long long idx = (long long)blockIdx.x * blockDim.x + threadIdx.x;
    if (idx >= (long long)E * 3) return;
    int e = (int)(idx / 3);
    int c = (int)(idx - (long long)e * 3);
    atomicAdd(&acc3[(size_t)dst[e] * 3 + c], hws3[(size_t)src[e] * 3 + c]);
}

__global__ void k_final_out(const float* __restrict__ acc3, const float* __restrict__ hws3,
                            const float* __restrict__ dinv, const float* __restrict__ b3,
                            const float* __restrict__ x, const float* __restrict__ pos,
                            float* __restrict__ out, int N) {
    int n = blockIdx.x * blockDim.x + threadIdx.x;
    if (n >= N) return;
    float di = dinv[n];
    float o0 = di * (acc3[n * 3 + 0] + hws3[n * 3 + 0]) + b3[0];
    float o1 = di * (acc3[n * 3 + 1] + hws3[n * 3 + 1]) + b3[1];
    float o2 = di * (acc3[n * 3 + 2] + hws3[n * 3 + 2]) + b3[2];
    out[n] = x[n] + o2;                       // new_type  [N,1]
    out[N + n * 2 + 0] = pos[n * 2 + 0] + o0; // new_pos   [N,2]
    out[N + n * 2 + 1] = pos[n * 2 + 1] + o1;
}

// ---------------------------------------------------------------------------
// Launch
// ---------------------------------------------------------------------------
extern "C" void kernel_launch(void* const* d_in, const int* in_sizes, int n_in,
                              void* d_out, int out_size, void* d_ws, size_t ws_size,
                              hipStream_t stream) {
    (void)n_in; (void)out_size; (void)ws_size;

    const float* x     = (const float*)d_in[0];   // [N,1]
    const float* pos   = (const float*)d_in[1];   // [N,2]
    const float* delta = (const float*)d_in[2];   // [4]
    const int*   ei    = (const int*)d_in[3];     // [2,E]
    const float* W1    = (const float*)d_in[4];   // [5,64]
    const float* b1    = (const float*)d_in[5];   // [64]
    const float* W2    = (const float*)d_in[6];   // [64,64]
    const float* b2    = (const float*)d_in[7];   // [64]
    const float* W3    = (const float*)d_in[8];   // [64,3]
    const float* b3    = (const float*)d_in[9];   // [3]

    const int N = in_sizes[0];
    const int E = in_sizes[3] / 2;
    const int* src = ei;
    const int* dst = ei + E;

    float* ws   = (float*)d_ws;
    float* dinv = ws;                       // [N]      (also used for degree)
    float* hws  = dinv + N;                 // [N,64]
    float* acc  = hws + (size_t)N * HID;    // [N,64]
    float* h    = acc + (size_t)N * HID;    // [N,64]

    const int T = 256;
    const int gN   = (N + T - 1) / T;
    const int gE   = (E + T - 1) / T;
    const int gNH  = (N * HID + T - 1) / T;
    const int gE16 = (int)(((long long)E * 16 + T - 1) / T);
    const int gE3  = (int)(((long long)E * 3 + T - 1) / T);
    const int gN3  = (N * 3 + T - 1) / T;
    const int gGemm = (((N + 15) / 16) + 7) / 8;   // 8 waves (16-row tiles) per block

    // Normalization
    k_init_deg<<<gN, T, 0, stream>>>(dinv, N);
    k_deg_edges<<<gE, T, 0, stream>>>(dinv, dst, E);
    k_deg_to_dinv<<<gN, T, 0, stream>>>(dinv, N);

    // ---- Layer 1 ----
    k_l1_hws<<<gNH, T, 0, stream>>>(x, delta, W1, dinv, hws, N);
    hipMemsetAsync(acc, 0, (size_t)N * HID * sizeof(float), stream);
    k_scatter64<<<gE16, T, 0, stream>>>(hws, src, dst, acc, E);
    k_finalize64<<<gNH, T, 0, stream>>>(acc, hws, dinv, b1, h, N, 1);

    // ---- Layer 2 (WMMA GEMM) ----
    k_gemm64_wmma<<<gGemm, T, 0, stream>>>(h, W2, dinv, hws, N);
    hipMemsetAsync(acc, 0, (size_t)N * HID * sizeof(float), stream);
    k_scatter64<<<gE16, T, 0, stream>>>(hws, src, dst, acc, E);
    k_finalize64<<<gNH, T, 0, stream>>>(acc, hws, dinv, b2, h, N, 1);

    // ---- Layer 3 ----
    k_l3_hws<<<gN3, T, 0, stream>>>(h, W3, dinv, hws, N);      // hws reused as [N,3]
    hipMemsetAsync(acc, 0, (size_t)N * 3 * sizeof(float), stream);
    k_scatter3<<<gE3, T, 0, stream>>>(hws, src, dst, acc, E);  // acc reused as [N,3]
    k_final_out<<<gN, T, 0, stream>>>(acc, hws, dinv, b3, x, pos, (float*)d_out, N);
}